// SparseVariableRouter_46179488367239
// MI455X (gfx1250) — compile-verified
//
#include <hip/hip_runtime.h>
#include <hip/hip_bf16.h>
#include <math.h>

// Problem constants (from reference): B=8, L=512, N=2048, H=16
#define B_   8
#define L_   512
#define N_   2048
#define H_   16
#define KSTAT 90      // max(32, int(sqrt(2048)*2))
#define KDYN  33      // max(16, int(log2(2048)*3))

typedef float v2f __attribute__((ext_vector_type(2)));
typedef float v8f __attribute__((ext_vector_type(8)));

// ---------------- workspace layout (bytes) ----------------
static const size_t OFF_NORMED = 0;                        // 4  MiB  (L*N f32)
static const size_t OFF_SIM    = 4ull  << 20;              // 16 MiB  (N*N f32)
static const size_t OFF_SIMW   = 20ull << 20;              // 16 MiB  (N*N f32)
static const size_t OFF_ROUTED = OFF_SIM;                  // 32 MiB  (B*N*L f32) reuses sim+simw
static const size_t OFF_HIST   = 36ull << 20;              // 256 KiB (65536 u32)
static const size_t OFF_LHIST  = OFF_HIST + 65536ull * 4;  // 1 MiB   (4*65536 u32)
static const size_t OFF_SEL    = OFF_LHIST + 4ull * 65536 * 4; // 32 B (4*(bin,rank))
static const size_t OFF_LOHI   = OFF_SEL + 256;            // 8 B (lo,hi)
static const size_t OFF_SIDX   = OFF_LOHI + 256;           // 768 KiB (N*96 i32)
static const size_t OFF_XT     = 39ull << 20;              // 32 MiB  (B*N*L f32)
static const size_t OFF_Q      = 71ull << 20;              // 1 MiB   (B*N*16 f32)
static const size_t OFF_K      = 72ull << 20;              // 1 MiB
static const size_t OFF_IMP    = 73ull << 20;              // 64 KiB  (B*N f32)

// sortable-uint transform: ascending key order == ascending float order
__device__ __forceinline__ unsigned int fkey(float f) {
  unsigned int u = __float_as_uint(f);
  return (u & 0x80000000u) ? ~u : (u | 0x80000000u);
}
__device__ __forceinline__ float funkey(unsigned int key) {
  unsigned int u = (key & 0x80000000u) ? (key & 0x7FFFFFFFu) : ~key;
  return __uint_as_float(u);
}

// ---- 1. normed[l,n] = mean_b(x)/ (std_b(x,ddof=1)+1e-5) ----
__global__ void k_normed(const float* __restrict__ x, float* __restrict__ nm) {
  int i = blockIdx.x * blockDim.x + threadIdx.x;      // over L*N, coalesced in n
  float s = 0.f, s2 = 0.f;
  for (int b = 0; b < B_; ++b) {
    float v = x[b * (L_ * N_) + i];
    s += v; s2 += v * v;
  }
  float mean = s * (1.0f / B_);
  float var  = (s2 - (float)B_ * mean * mean) * (1.0f / (B_ - 1));
  float sd   = sqrtf(fmaxf(var, 0.f)) + 1e-5f;
  nm[i] = mean / sd;
}

// ---- 2. sim = normed^T @ normed  via V_WMMA_F32_16X16X4_F32 ----
// One wave per 16x16 output tile; 4 waves per block; K=512 in steps of 4.
__global__ void k_sim_wmma(const float* __restrict__ nm, float* __restrict__ sim) {
  int lane = threadIdx.x & 31;
  int wid  = threadIdx.x >> 5;
  int tile = blockIdx.x * 4 + wid;        // 128*128 = 16384 tiles
  int i0 = (tile >> 7) << 4;
  int j0 = (tile & 127) << 4;
  int m  = lane & 15;
  int kh = (lane >> 4) << 1;              // lanes 0-15 hold K={0,1}, lanes 16-31 K={2,3}
  v8f acc = {};
  for (int k0 = 0; k0 < L_; k0 += 4) {
    v2f a, b;
    a.x = nm[(k0 + kh    ) * N_ + i0 + m];
    a.y = nm[(k0 + kh + 1) * N_ + i0 + m];
    b.x = nm[(k0 + kh    ) * N_ + j0 + m];
    b.y = nm[(k0 + kh + 1) * N_ + j0 + m];
    acc = __builtin_amdgcn_wmma_f32_16x16x4_f32(
        /*neg_a=*/false, a, /*neg_b=*/false, b,
        /*c_mod=*/(short)0, acc, /*reuse_a=*/false, /*reuse_b=*/false);
  }
  // C/D layout: VGPR v -> M = v + 8*(lane>=16), N = lane&15
  int half = lane >> 4;
  int col  = lane & 15;
#pragma unroll
  for (int v = 0; v < 8; ++v) {
    int row = v + 8 * half;
    sim[(size_t)(i0 + row) * N_ + j0 + col] = acc[v];
  }
}

// ---- 3. zero histogram region ----
__global__ void k_zero(unsigned int* __restrict__ p, int n) {
  int i = blockIdx.x * blockDim.x + threadIdx.x;
  if (i < n) p[i] = 0u;
}

// ---- 4. level-1 histogram on top 16 key bits ----
__global__ void k_hist1(const float* __restrict__ sim, unsigned int* __restrict__ hist) {
  int stride = gridDim.x * blockDim.x;
  for (int i = blockIdx.x * blockDim.x + threadIdx.x; i < N_ * N_; i += stride)
    atomicAdd(&hist[fkey(sim[i]) >> 16], 1u);
}

// ---- 5. locate bins for the 4 order statistics ----
__global__ void k_select1(const unsigned int* __restrict__ hist, int* __restrict__ sel) {
  if (threadIdx.x == 0 && blockIdx.x == 0) {
    // ranks: floor/ceil of 0.05*(N*N-1) and 0.95*(N*N-1)
    const long long ranks[4] = {209715LL, 209716LL, 3984587LL, 3984588LL};
    long long cum = 0; int t = 0;
    for (int bin = 0; bin < 65536 && t < 4; ++bin) {
      long long c = (long long)hist[bin];
      while (t < 4 && ranks[t] < cum + c) {
        sel[2 * t]     = bin;
        sel[2 * t + 1] = (int)(ranks[t] - cum);
        ++t;
      }
      cum += c;
    }
  }
}

// ---- 6. level-2 histograms on low 16 bits, restricted per target bin ----
__global__ void k_hist2(const float* __restrict__ sim, const int* __restrict__ sel,
                        unsigned int* __restrict__ lhist) {
  int stride = gridDim.x * blockDim.x;
  int b0 = sel[0], b1 = sel[2], b2 = sel[4], b3 = sel[6];
  for (int i = blockIdx.x * blockDim.x + threadIdx.x; i < N_ * N_; i += stride) {
    unsigned int key = fkey(sim[i]);
    int hi = (int)(key >> 16);
    unsigned int lo = key & 0xFFFFu;
    if (hi == b0) atomicAdd(&lhist[0 * 65536 + lo], 1u);
    if (hi == b1) atomicAdd(&lhist[1 * 65536 + lo], 1u);
    if (hi == b2) atomicAdd(&lhist[2 * 65536 + lo], 1u);
    if (hi == b3) atomicAdd(&lhist[3 * 65536 + lo], 1u);
  }
}

// ---- 7. finish exact selection + linear interpolation -> lo,hi ----
__global__ void k_select2(const unsigned int* __restrict__ lhist,
                          const int* __restrict__ sel, float* __restrict__ lohi) {
  if (threadIdx.x == 0 && blockIdx.x == 0) {
    float v[4];
    for (int t = 0; t < 4; ++t) {
      long long rank = sel[2 * t + 1];
      unsigned int bin = (unsigned int)sel[2 * t];
      long long cum = 0; unsigned int low = 0;
      for (unsigned int j = 0; j < 65536u; ++j) {
        cum += (long long)lhist[t * 65536 + j];
        if (cum > rank) { low = j; break; }
      }
      v[t] = funkey((bin << 16) | low);
    }
    lohi[0] = v[0] + 0.15f * (v[1] - v[0]);   // 5% quantile
    lohi[1] = v[2] + 0.85f * (v[3] - v[2]);   // 95% quantile
  }
}

// ---- 8. winsorize + symmetrize + zero-diag + relu ----
__global__ void k_winsor(const float* __restrict__ sim, const float* __restrict__ lohi,
                         float* __restrict__ simw) {
  int i = blockIdx.x * blockDim.x + threadIdx.x;     // over N*N
  int row = i >> 11, col = i & (N_ - 1);
  float lo = lohi[0], hi = lohi[1];
  float a = fminf(fmaxf(sim[i], lo), hi);
  float b = fminf(fmaxf(sim[(size_t)col * N_ + row], lo), hi);
  float s = 0.5f * (a + b);
  if (row == col) s = 0.f;
  simw[i] = fmaxf(s, 0.f);
}

// ---- 9. per-row top-90 indices (exact, tie-break by lower index) ----
__global__ void k_rowtopk(const float* __restrict__ simw, int* __restrict__ sidx) {
  __shared__ float vals[N_];
  __shared__ float rv[256];
  __shared__ int   ri[256];
  int row = blockIdx.x;
  for (int c = threadIdx.x; c < N_; c += 256) vals[c] = simw[(size_t)row * N_ + c];
  __syncthreads();
  for (int it = 0; it < KSTAT; ++it) {
    float bv = -3e38f; int bi = 0;
    for (int c = threadIdx.x; c < N_; c += 256) {
      float v = vals[c];
      if (v > bv) { bv = v; bi = c; }
    }
    rv[threadIdx.x] = bv; ri[threadIdx.x] = bi;
    __syncthreads();
    for (int s = 128; s > 0; s >>= 1) {
      if ((int)threadIdx.x < s) {
        float ov = rv[threadIdx.x + s]; int oi = ri[threadIdx.x + s];
        if (ov > rv[threadIdx.x] || (ov == rv[threadIdx.x] && oi < ri[threadIdx.x])) {
          rv[threadIdx.x] = ov; ri[threadIdx.x] = oi;
        }
      }
      __syncthreads();
    }
    if (threadIdx.x == 0) { sidx[row * 96 + it] = ri[0]; vals[ri[0]] = -3e38f; }
    __syncthreads();
  }
}

// ---- 10. transpose x (B,L,N) -> xt (B,N,L) ----
__global__ void k_transpose(const float* __restrict__ x, float* __restrict__ xt) {
  __shared__ float t[32][33];
  int b = blockIdx.z;
  int n0 = blockIdx.x * 32, l0 = blockIdx.y * 32;
  int tx = threadIdx.x, ty = threadIdx.y;   // (32,8)
  for (int r = 0; r < 32; r += 8)
    t[ty + r][tx] = x[(size_t)b * L_ * N_ + (size_t)(l0 + ty + r) * N_ + n0 + tx];
  __syncthreads();
  for (int r = 0; r < 32; r += 8)
    xt[(size_t)b * N_ * L_ + (size_t)(n0 + ty + r) * L_ + l0 + tx] = t[tx][ty + r];
}

// ---- 11. per-(b,n): time stats, fused features, Q/K, importance ----
__global__ void k_feat(const float* __restrict__ xt, const float* __restrict__ vemb,
                       const float* __restrict__ Wt, const float* __restrict__ bt,
                       const float* __restrict__ Wq, const float* __restrict__ bq,
                       const float* __restrict__ Wk, const float* __restrict__ bk,
                       const float* __restrict__ Wg, const float* __restrict__ bg,
                       float* __restrict__ Q, float* __restrict__ K, float* __restrict__ imp) {
  __shared__ float fused[8][32];
  int lane = threadIdx.x & 31, wid = threadIdx.x >> 5;
  int row = blockIdx.x * 8 + wid;           // row = b*N + n
  int n = row & (N_ - 1);
  const float* xr = xt + (size_t)row * L_;
  float s = 0.f, s2 = 0.f;
  for (int l = lane; l < L_; l += 32) { float v = xr[l]; s += v; s2 += v * v; }
  for (int o = 16; o > 0; o >>= 1) { s += __shfl_xor(s, o, 32); s2 += __shfl_xor(s2, o, 32); }
  float mean = s * (1.0f / L_);
  float var  = (s2 - (float)L_ * mean * mean) * (1.0f / (L_ - 1));
  float sd   = sqrtf(fmaxf(var, 0.f) + 1e-5f);
  if (lane < 16) {
    fused[wid][lane] = vemb[n * H_ + lane];                       // static embedding
  } else {
    int h = lane - 16;
    fused[wid][16 + h] = Wt[h * 2 + 0] * mean + Wt[h * 2 + 1] * sd + bt[h];  // time emb
  }
  __syncthreads();
  int h = lane & 15;
  const float* W = (lane < 16) ? Wq : Wk;
  float acc = (lane < 16) ? bq[h] : bk[h];
  for (int c = 0; c < 32; ++c) acc += fused[wid][c] * W[h * 32 + c];
  if (lane < 16) Q[(size_t)row * 16 + h] = acc;
  else           K[(size_t)row * 16 + h] = acc;
  if (lane == 0) {
    float g = bg[0];
    for (int c = 0; c < 32; ++c) g += fused[wid][c] * Wg[c];
    imp[row] = 1.f / (1.f + expf(-g));
  }
}

// ---- 12. sparse route: per row gather 90 candidates, top-33, softmax, 33-term FMA ----
__global__ void k_route(const float* __restrict__ xt, const float* __restrict__ Q,
                        const float* __restrict__ K, const int* __restrict__ sidx,
                        float* __restrict__ routed) {
  __shared__ float cval[8][KSTAT];
  __shared__ int   cidx[8][KSTAT];
  __shared__ float wgt[8][KDYN];
  __shared__ int   wjx[8][KDYN];
  int lane = threadIdx.x & 31, wid = threadIdx.x >> 5;
  int row = blockIdx.x * 8 + wid;           // row = b*N + n
  int b = row >> 11, n = row & (N_ - 1);
  float q[16];
#pragma unroll
  for (int h = 0; h < 16; ++h) q[h] = Q[(size_t)row * 16 + h];
  for (int c = lane; c < KSTAT; c += 32) {
    int j = sidx[n * 96 + c];
    const float* kr = K + ((size_t)b * N_ + j) * 16;
    float d = 0.f;
#pragma unroll
    for (int h = 0; h < 16; ++h) d += q[h] * kr[h];
    cval[wid][c] = d * 0.25f;               // 1/sqrt(H)
    cidx[wid][c] = j;
  }
  __syncthreads();
  // top-33 of 90 by value (tie-break lower column index), wave-parallel argmax
  for (int it = 0; it < KDYN; ++it) {
    float bv = -3e38f; int bj = 0x7fffffff; int bs = 0;
    for (int c = lane; c < KSTAT; c += 32) {
      float v = cval[wid][c]; int j = cidx[wid][c];
      if (v > bv || (v == bv && j < bj)) { bv = v; bj = j; bs = c; }
    }
    for (int o = 16; o > 0; o >>= 1) {
      float ov = __shfl_xor(bv, o, 32);
      int   oj = __shfl_xor(bj, o, 32);
      int   os = __shfl_xor(bs, o, 32);
      if (ov > bv || (ov == bv && oj < bj)) { bv = ov; bj = oj; bs = os; }
    }
    if (lane == 0) { wgt[wid][it] = bv; wjx[wid][it] = bj; cval[wid][bs] = -3e38f; }
    __syncthreads();
  }
  // softmax over the 33 survivors (max is the first selected)
  float m = wgt[wid][0];
  float e0 = (lane < KDYN) ? expf(wgt[wid][lane] - m) : 0.f;
  float e1 = (lane == 0) ? expf(wgt[wid][KDYN - 1] - m) : 0.f;
  float ssum = ((lane < KDYN - 1) ? e0 : 0.f) + e1;
  // note: lane KDYN-1==32 doesn't exist in wave32; handle index 32 via lane 0 (e1)
  ssum = ((lane < 32 && lane < KDYN) ? e0 : 0.f);
  if (lane == 0) ssum += e1 - e0;  // lane0 contributes e(idx0)+e(idx32)
  if (lane == 0) ssum += e0;
  for (int o = 16; o > 0; o >>= 1) ssum += __shfl_xor(ssum, o, 32);
  float inv = 1.f / ssum;
  if (lane < 32 && lane < KDYN) wgt[wid][lane] = e0 * inv;
  if (lane == 0) wgt[wid][KDYN - 1] = e1 * inv;
  __syncthreads();
  // routed[b,n,:] = sum_t w_t * xt[b, j_t, :]   (coalesced 32-lane chunks)
  float* out = routed + (size_t)row * L_;
  for (int l0 = 0; l0 < L_; l0 += 32) {
    float acc = 0.f;
#pragma unroll 1
    for (int t = 0; t < KDYN; ++t) {
      const float* xr = xt + ((size_t)b * N_ + wjx[wid][t]) * L_;
      if (l0 + 32 < L_) __builtin_prefetch(xr + l0 + 32, 0, 1);   // global_prefetch_b8
      acc += wgt[wid][t] * xr[l0 + lane];
    }
    out[l0 + lane] = acc;
  }
}

// ---- 13. out[b,l,n] = imp*routed[b,n,l] + (1-imp)*x[b,l,n]  (tiled transpose) ----
__global__ void k_combine(const float* __restrict__ x, const float* __restrict__ routed,
                          const float* __restrict__ imp, float* __restrict__ out) {
  __shared__ float t[32][33];
  int b = blockIdx.z;
  int n0 = blockIdx.x * 32, l0 = blockIdx.y * 32;
  int tx = threadIdx.x, ty = threadIdx.y;   // (32,8)
  for (int r = 0; r < 32; r += 8)
    t[ty + r][tx] = routed[((size_t)b * N_ + n0 + ty + r) * L_ + l0 + tx];
  __syncthreads();
  for (int r = 0; r < 32; r += 8) {
    int l = l0 + ty + r, n = n0 + tx;
    float ip = imp[b * N_ + n];
    size_t o = (size_t)b * L_ * N_ + (size_t)l * N_ + n;
    out[o] = ip * t[tx][ty + r] + (1.f - ip) * x[o];
  }
}

extern "C" void kernel_launch(void* const* d_in, const int* in_sizes, int n_in,
                              void* d_out, int out_size, void* d_ws, size_t ws_size,
                              hipStream_t stream) {
  (void)in_sizes; (void)n_in; (void)out_size; (void)ws_size;
  const float* x    = (const float*)d_in[0];
  const float* vemb = (const float*)d_in[1];
  const float* Wt   = (const float*)d_in[2];
  const float* bt   = (const float*)d_in[3];
  const float* Wq   = (const float*)d_in[4];
  const float* bq   = (const float*)d_in[5];
  const float* Wk   = (const float*)d_in[6];
  const float* bk   = (const float*)d_in[7];
  const float* Wg   = (const float*)d_in[8];
  const float* bg   = (const float*)d_in[9];
  float* out = (float*)d_out;
  char* ws = (char*)d_ws;

  float*        nm    = (float*)(ws + OFF_NORMED);
  float*        sim   = (float*)(ws + OFF_SIM);
  float*        simw  = (float*)(ws + OFF_SIMW);
  float*        routed= (float*)(ws + OFF_ROUTED);
  unsigned int* hist  = (unsigned int*)(ws + OFF_HIST);
  unsigned int* lhist = (unsigned int*)(ws + OFF_LHIST);
  int*          sel   = (int*)(ws + OFF_SEL);
  float*        lohi  = (float*)(ws + OFF_LOHI);
  int*          sidx  = (int*)(ws + OFF_SIDX);
  float*        xt    = (float*)(ws + OFF_XT);
  float*        Qb    = (float*)(ws + OFF_Q);
  float*        Kb    = (float*)(ws + OFF_K);
  float*        impb  = (float*)(ws + OFF_IMP);

  // Stage A: static mask
  k_normed  <<<(L_ * N_) / 256, 256, 0, stream>>>(x, nm);
  k_sim_wmma<<<(N_ / 16) * (N_ / 16) / 4, 128, 0, stream>>>(nm, sim);
  int nz = 5 * 65536 + 128;
  k_zero    <<<(nz + 255) / 256, 256, 0, stream>>>(hist, nz);
  k_hist1   <<<4096, 256, 0, stream>>>(sim, hist);
  k_select1 <<<1, 32, 0, stream>>>(hist, sel);
  k_hist2   <<<4096, 256, 0, stream>>>(sim, sel, lhist);
  k_select2 <<<1, 32, 0, stream>>>(lhist, sel, lohi);
  k_winsor  <<<(N_ * N_) / 256, 256, 0, stream>>>(sim, lohi, simw);
  k_rowtopk <<<N_, 256, 0, stream>>>(simw, sidx);

  // Stage B: dynamic routing (routed overwrites sim/simw region — safe, stream-ordered)
  k_transpose<<<dim3(N_ / 32, L_ / 32, B_), dim3(32, 8), 0, stream>>>(x, xt);
  k_feat     <<<(B_ * N_) / 8, 256, 0, stream>>>(xt, vemb, Wt, bt, Wq, bq, Wk, bk, Wg, bg,
                                                 Qb, Kb, impb);
  k_route    <<<(B_ * N_) / 8, 256, 0, stream>>>(xt, Qb, Kb, sidx, routed);
  k_combine  <<<dim3(N_ / 32, L_ / 32, B_), dim3(32, 8), 0, stream>>>(x, routed, impb, out);
}